// Decoder_29051158790190
// MI455X (gfx1250) — compile-verified
//
#include <hip/hip_runtime.h>
#include <math.h>

// Problem constants (from reference)
#define BB   32
#define TT   1500
#define EE   512
#define HH   512
#define AA   512
#define VV   10000
#define LL   100
#define LP1  101
#define G4H  2048     // 4*H
#define KXA  1024     // E + A (gate GEMM phase-1 K)
#define KZC  1024     // H + A (output GEMM K)

typedef __attribute__((ext_vector_type(2))) float v2f;
typedef __attribute__((ext_vector_type(8))) float v8f;

static __device__ __forceinline__ v8f wmma4(v2f a, v2f b, v8f c) {
  // D = A(16x4) * B(4x16) + C(16x16), fp32, wave32
  return __builtin_amdgcn_wmma_f32_16x16x4_f32(false, a, false, b, (short)0, c,
                                               false, false);
}

static __device__ __forceinline__ float sigmoidf_(float x) {
  return 1.0f / (1.0f + expf(-x));
}

// CDNA5 native transcendental tanh (TRANS32). s_nop covers the
// trans-result-use hazard (1 independent op required before use).
static __device__ __forceinline__ float fast_tanh(float x) {
  float y;
  asm volatile("v_tanh_f32 %0, %1\n\ts_nop 1" : "=v"(y) : "v"(x));
  return y;
}

// ---------------------------------------------------------------------------
// fp32 WMMA GEMM, compile-time specialized:
//   TRANSB: B is [N,K] row-major (1) or [K,N] row-major (0)
//   MT:     M-tiles per wave (1: streaming-M; 2: M=32 GEMMs share B-fragment)
//   HASBIAS: add bias[n]
// One wave per 16-wide N tile; branchless inner loop so loads clause/pipeline
// across the WMMA accumulator chain.
// ---------------------------------------------------------------------------
template <int TRANSB, int MT, int HASBIAS>
__global__ void wmma_gemm_t(const float* __restrict__ A, int lda,
                            const float* __restrict__ Bm, int ldb,
                            float* __restrict__ C, size_t ldc,
                            int Ntiles, int K,
                            const float* __restrict__ bias) {
  int wv   = threadIdx.x >> 5;
  int lane = threadIdx.x & 31;
  int ntile = blockIdx.x * 4 + wv;   // wave-uniform guard: EXEC stays all-1s
  if (ntile >= Ntiles) return;
  int mtile0 = blockIdx.y * MT;

  int mr   = lane & 15;
  int koff = (lane >> 4) << 1;       // lanes 0-15 -> K 0,1 ; lanes 16-31 -> K 2,3
  int n    = ntile * 16 + mr;

  v8f acc0 = {0.f, 0.f, 0.f, 0.f, 0.f, 0.f, 0.f, 0.f};
  v8f acc1 = {0.f, 0.f, 0.f, 0.f, 0.f, 0.f, 0.f, 0.f};

#pragma unroll 2
  for (int k = 0; k < K; k += 4) {
    v2f bv;
    if (TRANSB) {
      const float* bp = Bm + (size_t)n * ldb + (k + koff);
      bv.x = bp[0]; bv.y = bp[1];
    } else {
      bv.x = Bm[(size_t)(k + koff) * ldb + n];
      bv.y = Bm[(size_t)(k + koff + 1) * ldb + n];
    }
    const float* ap0 = A + (size_t)(mtile0 * 16 + mr) * lda + (k + koff);
    v2f a0; a0.x = ap0[0]; a0.y = ap0[1];
    acc0 = wmma4(a0, bv, acc0);
    if (MT == 2) {
      const float* ap1 = ap0 + (size_t)16 * lda;
      v2f a1; a1.x = ap1[0]; a1.y = ap1[1];
      acc1 = wmma4(a1, bv, acc1);
    }
  }

  float bval = HASBIAS ? bias[n] : 0.0f;
  int rb = (lane >> 4) << 3;
#pragma unroll
  for (int r = 0; r < 8; ++r) {
    C[(size_t)(mtile0 * 16 + rb + r) * ldc + n] = acc0[r] + bval;
    if (MT == 2)
      C[(size_t)((mtile0 + 1) * 16 + rb + r) * ldc + n] = acc1[r] + bval;
  }
}

// ---------------------------------------------------------------------------
// LSTM gates: gates[32,2048] = X[32,1024]@W_ih^T + Z[32,512]@W_hh^T + b_ih+b_hh
// Two fused K-phases; both 16-row M tiles per wave share the B fragment.
// ---------------------------------------------------------------------------
__global__ void gates_wmma_kernel(const float* __restrict__ X,
                                  const float* __restrict__ Wih,
                                  const float* __restrict__ Z,
                                  const float* __restrict__ Whh,
                                  const float* __restrict__ bih,
                                  const float* __restrict__ bhh,
                                  float* __restrict__ gates) {
  int wv   = threadIdx.x >> 5;
  int lane = threadIdx.x & 31;
  int ntile = blockIdx.x * 4 + wv;   // 0..127
  int mr   = lane & 15;
  int koff = (lane >> 4) << 1;
  int n    = ntile * 16 + mr;

  v8f acc0 = {0.f, 0.f, 0.f, 0.f, 0.f, 0.f, 0.f, 0.f};
  v8f acc1 = {0.f, 0.f, 0.f, 0.f, 0.f, 0.f, 0.f, 0.f};

#pragma unroll 2
  for (int k = 0; k < KXA; k += 4) {
    const float* bp = Wih + (size_t)n * KXA + k + koff;
    v2f bv; bv.x = bp[0]; bv.y = bp[1];
    const float* ap0 = X + (size_t)mr * KXA + k + koff;
    v2f a0; a0.x = ap0[0]; a0.y = ap0[1];
    acc0 = wmma4(a0, bv, acc0);
    const float* ap1 = ap0 + (size_t)16 * KXA;
    v2f a1; a1.x = ap1[0]; a1.y = ap1[1];
    acc1 = wmma4(a1, bv, acc1);
  }
#pragma unroll 2
  for (int k = 0; k < HH; k += 4) {
    const float* bp = Whh + (size_t)n * HH + k + koff;
    v2f bv; bv.x = bp[0]; bv.y = bp[1];
    const float* ap0 = Z + (size_t)mr * HH + k + koff;
    v2f a0; a0.x = ap0[0]; a0.y = ap0[1];
    acc0 = wmma4(a0, bv, acc0);
    const float* ap1 = ap0 + (size_t)16 * HH;
    v2f a1; a1.x = ap1[0]; a1.y = ap1[1];
    acc1 = wmma4(a1, bv, acc1);
  }

  float bval = bih[n] + bhh[n];
  int rb = (lane >> 4) << 3;
#pragma unroll
  for (int r = 0; r < 8; ++r) {
    gates[(size_t)(rb + r) * G4H + n]      = acc0[r] + bval;
    gates[(size_t)(16 + rb + r) * G4H + n] = acc1[r] + bval;
  }
}

// ---------------------------------------------------------------------------
// Small helper kernels
// ---------------------------------------------------------------------------
__global__ void zero_kernel(float* __restrict__ p, int n) {
  int i = blockIdx.x * blockDim.x + threadIdx.x;
  if (i < n) p[i] = 0.f;
}

// eys[(t*32+b)*512 + :] = emb_table[ys_in[b,t]]
__global__ void embed_kernel(const int* __restrict__ ys,
                             const float* __restrict__ emb,
                             float* __restrict__ eys) {
  int blk = blockIdx.x;            // t*32 + b
  int t = blk >> 5;
  int b = blk & 31;
  int tok = (t == 0) ? 1 /*BOS*/ : ys[b * LL + (t - 1)];
  const float* src = emb + (size_t)tok * EE;
  float* dst = eys + (size_t)blk * EE;
  for (int i = threadIdx.x; i < EE; i += blockDim.x) dst[i] = src[i];
}

// out[32,1024] = [a[32,512] | b[32,512]]
__global__ void concat_kernel(const float* __restrict__ a,
                              const float* __restrict__ b,
                              float* __restrict__ out) {
  int i = blockIdx.x * 256 + threadIdx.x;    // 0..32767
  int row = i >> 10, col = i & 1023;
  out[i] = (col < 512) ? a[row * 512 + col] : b[row * 512 + (col - 512)];
}

__global__ void lstm_kernel(const float* __restrict__ gates,
                            float* __restrict__ z, float* __restrict__ c) {
  int i = blockIdx.x * 512 + threadIdx.x;    // 0..16383
  int b = i >> 9, h = i & 511;
  const float* g = gates + (size_t)b * G4H;
  float ig = sigmoidf_(g[h]);
  float fg = sigmoidf_(g[512 + h]);
  float gg = tanhf(g[1024 + h]);
  float og = sigmoidf_(g[1536 + h]);
  float cn = fg * c[i] + ig * gg;
  c[i] = cn;
  z[i] = og * tanhf(cn);
}

// e[b,t] = v_att . tanh(enc_proj[b,t,:] + s[b,:]), masked by enc_len.
// One wave per (b,t); lane-strided coalesced loads + shfl_xor reduction.
// Uses native v_tanh_f32 (TRANS32) for the 24.5M tanh/step.
__global__ void att_score_kernel(const float* __restrict__ enc_proj,
                                 const float* __restrict__ s,
                                 const float* __restrict__ v_att,
                                 const int* __restrict__ enc_len,
                                 float* __restrict__ e) {
  int gw = (blockIdx.x * blockDim.x + threadIdx.x) >> 5;  // 0..47999 (grid exact)
  int lane = threadIdx.x & 31;
  int b = gw / TT;
  int t = gw - b * TT;
  const float* ep = enc_proj + (size_t)gw * AA;
  const float* sb = s + b * AA;
  float acc = 0.f;
#pragma unroll
  for (int j = 0; j < 16; ++j) {
    int idx = lane + j * 32;
    acc += fast_tanh(ep[idx] + sb[idx]) * v_att[idx];
  }
#pragma unroll
  for (int off = 16; off > 0; off >>= 1) acc += __shfl_xor(acc, off, 32);
  if (lane == 0) e[gw] = (t < enc_len[b]) ? acc : -1.0e9f;
}

// Per-b masked softmax over T; also writes attention weights into ws output.
__global__ void softmax_kernel(const float* __restrict__ e,
                               float* __restrict__ w,
                               float* __restrict__ ws_out, int step) {
  __shared__ float red[512];
  int b = blockIdx.x, tid = threadIdx.x;
  const float* eb = e + (size_t)b * TT;
  float m = -3.0e38f;
  for (int t = tid; t < TT; t += 512) m = fmaxf(m, eb[t]);
  red[tid] = m; __syncthreads();
  for (int s = 256; s > 0; s >>= 1) {
    if (tid < s) red[tid] = fmaxf(red[tid], red[tid + s]);
    __syncthreads();
  }
  m = red[0]; __syncthreads();
  float sum = 0.f;
  for (int t = tid; t < TT; t += 512) sum += expf(eb[t] - m);
  red[tid] = sum; __syncthreads();
  for (int s = 256; s > 0; s >>= 1) {
    if (tid < s) red[tid] += red[tid + s];
    __syncthreads();
  }
  float inv = 1.0f / red[0];
  for (int t = tid; t < TT; t += 512) {
    float wv = expf(eb[t] - m) * inv;
    w[(size_t)b * TT + t] = wv;
    ws_out[((size_t)b * LP1 + step) * TT + t] = wv;
  }
}

// ctx[b,d] = sum_t w[b,t] * enc_pad[b,t,d]  (coalesced over d)
__global__ void context_kernel(const float* __restrict__ w,
                               const float* __restrict__ enc,
                               float* __restrict__ ctx) {
  int b = blockIdx.x;
  int d = blockIdx.y * 128 + threadIdx.x;
  const float* wb = w + (size_t)b * TT;
  const float* eb = enc + (size_t)b * TT * AA + d;
  float acc = 0.f;
  for (int t = 0; t < TT; ++t) acc += wb[t] * eb[(size_t)t * AA];
  ctx[b * AA + d] = acc;
}

// Per-b: argmax + logsumexp over V; write ys_log_probs and prediction.
__global__ void logsm_kernel(const float* __restrict__ logits,
                             const int* __restrict__ ys,
                             float* __restrict__ ylp,
                             float* __restrict__ pred, int step) {
  __shared__ float sv[512];
  __shared__ int   si[512];
  int b = blockIdx.x, tid = threadIdx.x;
  const float* lg = logits + (size_t)b * LP1 * VV;  // row stride 101*V
  float bm = -3.0e38f; int bi = 0;
  for (int v = tid; v < VV; v += 512) {
    float x = lg[v];
    if (x > bm) { bm = x; bi = v; }
  }
  sv[tid] = bm; si[tid] = bi; __syncthreads();
  for (int s = 256; s > 0; s >>= 1) {
    if (tid < s) {
      float ov = sv[tid + s]; int oi = si[tid + s];
      if (ov > sv[tid] || (ov == sv[tid] && oi < si[tid])) {
        sv[tid] = ov; si[tid] = oi;
      }
    }
    __syncthreads();
  }
  float m = sv[0]; int am = si[0]; __syncthreads();
  float sum = 0.f;
  for (int v = tid; v < VV; v += 512) sum += expf(lg[v] - m);
  sv[tid] = sum; __syncthreads();
  for (int s = 256; s > 0; s >>= 1) {
    if (tid < s) sv[tid] += sv[tid + s];
    __syncthreads();
  }
  if (tid == 0) {
    float lse = m + logf(sv[0]);
    int tgt = (step < LL) ? ys[b * LL + step] : 2 /*EOS*/;
    ylp[b * LP1 + step]  = lg[tgt] - lse;
    pred[b * LP1 + step] = (float)am;
  }
}

// ---------------------------------------------------------------------------
extern "C" void kernel_launch(void* const* d_in, const int* in_sizes, int n_in,
                              void* d_out, int out_size, void* d_ws,
                              size_t ws_size, hipStream_t stream) {
  (void)in_sizes; (void)n_in; (void)out_size; (void)ws_size;

  const float* enc_pad = (const float*)d_in[0];
  const int*   enc_len = (const int*)d_in[1];
  const int*   ys      = (const int*)d_in[2];
  const float* emb     = (const float*)d_in[3];
  const float* W_ih    = (const float*)d_in[4];
  const float* W_hh    = (const float*)d_in[5];
  const float* b_ih    = (const float*)d_in[6];
  const float* b_hh    = (const float*)d_in[7];
  const float* W_enc   = (const float*)d_in[8];
  const float* W_dec   = (const float*)d_in[9];
  const float* v_att   = (const float*)d_in[10];
  const float* W_out   = (const float*)d_in[11];
  const float* b_out   = (const float*)d_in[12];

  // Output tuple layout: logits[B,101,V] | ys_log_probs[B,101] | pred[B,101] | ws[B,101,T]
  float* out        = (float*)d_out;
  float* out_logits = out;
  float* out_ylp    = out + (size_t)BB * LP1 * VV;
  float* out_pred   = out_ylp + BB * LP1;
  float* out_ws     = out_pred + BB * LP1;

  // Scratch layout (floats)
  float* ws = (float*)d_ws;
  size_t off = 0;
  float* enc_proj = ws + off; off += (size_t)BB * TT * AA;   // 24.58M
  float* eys      = ws + off; off += (size_t)LP1 * BB * EE;  // 1.65M
  float* xbuf     = ws + off; off += (size_t)BB * KXA;
  float* gbuf     = ws + off; off += (size_t)BB * G4H;
  float* zbuf     = ws + off; off += (size_t)BB * HH;        // contiguous with
  float* ccell    = ws + off; off += (size_t)BB * HH;        // ccell, ctx for
  float* ctx      = ws + off; off += (size_t)BB * AA;        // one zero pass
  float* satt     = ws + off; off += (size_t)BB * AA;
  float* ebuf     = ws + off; off += (size_t)BB * TT;
  float* wbuf     = ws + off; off += (size_t)BB * TT;

  // 1) zero initial LSTM state + context (zbuf/ccell/ctx are contiguous)
  zero_kernel<<<96, 512, 0, stream>>>(zbuf, 3 * BB * HH);

  // 2) embeddings for all steps (time-major)
  embed_kernel<<<LP1 * BB, 256, 0, stream>>>(ys, emb, eys);

  // 3) enc_proj = enc_pad @ W_enc  (M=48000, N=512, K=512) -- WMMA fp32
  wmma_gemm_t<0, 1, 0><<<dim3(8, 3000), 128, 0, stream>>>(
      enc_pad, AA, W_enc, 512, enc_proj, (size_t)512,
      /*Ntiles=*/32, /*K=*/512, nullptr);

  // 4) sequential decode, 101 steps
  for (int t = 0; t < LP1; ++t) {
    const float* emb_t = eys + (size_t)t * BB * EE;

    // cell_inp = [emb_t | ctx]
    concat_kernel<<<128, 256, 0, stream>>>(emb_t, ctx, xbuf);

    // gates = X@W_ih^T + z@W_hh^T + b_ih + b_hh   (WMMA, fused 2 phases)
    gates_wmma_kernel<<<dim3(32, 1), 128, 0, stream>>>(
        xbuf, W_ih, zbuf, W_hh, b_ih, b_hh, gbuf);

    // LSTM pointwise update of (z, c_cell)
    lstm_kernel<<<32, 512, 0, stream>>>(gbuf, zbuf, ccell);

    // s = z @ W_dec  (M=32, N=512, K=512) -- WMMA, 2 M-tiles/wave
    wmma_gemm_t<0, 2, 0><<<dim3(8, 1), 128, 0, stream>>>(
        zbuf, HH, W_dec, 512, satt, (size_t)512,
        /*Ntiles=*/32, /*K=*/512, nullptr);

    // attention scores (bandwidth-dominant: streams enc_proj 96MB)
    att_score_kernel<<<6000, 256, 0, stream>>>(enc_proj, satt, v_att,
                                               enc_len, ebuf);

    // masked softmax over T, also emits ws output slice
    softmax_kernel<<<32, 512, 0, stream>>>(ebuf, wbuf, out_ws, t);

    // context = w @ enc_pad (streams enc_pad 96MB)
    context_kernel<<<dim3(32, 4), 128, 0, stream>>>(wbuf, enc_pad, ctx);

    // logits = [z | ctx] @ W_out^T + b_out  (M=32, N=10000, K=1024)
    // WMMA, 2 M-tiles/wave -> W_out streamed once per step
    concat_kernel<<<128, 256, 0, stream>>>(zbuf, ctx, xbuf);
    wmma_gemm_t<1, 2, 1><<<dim3(157, 1), 128, 0, stream>>>(
        xbuf, KZC, W_out, KZC,
        out_logits + (size_t)t * VV, (size_t)LP1 * VV,
        /*Ntiles=*/625, /*K=*/KZC, b_out);

    // log-softmax gather + argmax
    logsm_kernel<<<32, 512, 0, stream>>>(out_logits + (size_t)t * VV, ys,
                                         out_ylp, out_pred, t);
  }
}